// MultiResolutionMamba_42279658062589
// MI455X (gfx1250) — compile-verified
//
#include <hip/hip_runtime.h>
#include <hip/hip_bf16.h>
#include <math.h>

// Problem constants (reference: B_, L_, DM, DS, DC, EXP = 2,4096,768,64,4,2)
#define B_  2
#define L_  4096
#define DM  768
#define DS  64
#define DC  4
#define DI  1536           // DM * EXP
#define BL  (B_ * L_)      // 8192 rows
#define NP  129            // 2*DS + 1 real xproj columns
#define PLD 144            // padded leading dim for p / xproj_w (9 tiles of 16)
#define CH  16             // timesteps per TDM-staged chunk in the scan

typedef __attribute__((ext_vector_type(2))) float        v2f;
typedef __attribute__((ext_vector_type(8))) float        v8f;
typedef __attribute__((ext_vector_type(4))) unsigned int u32x4;
typedef __attribute__((ext_vector_type(8))) int          i32x8;
typedef __attribute__((ext_vector_type(4))) int          i32x4;

__device__ __forceinline__ float silu_f(float x)     { return x / (1.0f + expf(-x)); }
__device__ __forceinline__ float softplus_f(float x) { return (x > 20.0f) ? x : log1pf(expf(x)); }

// ---------------------------------------------------------------------------
// TDM: async-load a CH x PLD fp32 tile (rows contiguous, stride PLD) from
// global memory into LDS at byte offset lds_off. Descriptor per ISA 8.3/8.4:
//   group0: count=1 | lds_addr | global_addr | type=2
//   group1: data_size=4B, tensor_dim0=PLD, tensor_dim1=BL, tile_dim0=PLD,
//           tile_dim1=CH, tensor_dim0_stride=PLD, workgroup_mask=0
// Tracked by TENSORcnt; caller pairs with s_wait_tensorcnt.
// ---------------------------------------------------------------------------
__device__ __forceinline__ void tdm_load_rows(const float* gptr, unsigned lds_off)
{
    const unsigned long long ga = (unsigned long long)(uintptr_t)gptr;
    u32x4 g0;
    g0.x = 1u;                                                  // count=1, normal mode
    g0.y = lds_off;                                             // LDS byte address
    g0.z = (unsigned)(ga & 0xFFFFFFFFull);                      // global addr [31:0]
    g0.w = (unsigned)((ga >> 32) & 0x01FFFFFFull) | 0x80000000u;// [56:32] | type=2<<30
    i32x8 g1;
    g1[0] = 0x20000;                          // data_size=2 (4 bytes); mask=0
    g1[1] = (int)((unsigned)PLD << 16);       // tensor_dim0[15:0] << 16
    g1[2] = (int)((unsigned)BL  << 16);       // tensor_dim0[31:16]=0 | tensor_dim1[15:0]
    g1[3] = (int)((unsigned)PLD << 16);       // tensor_dim1[31:16]=0 | tile_dim0
    g1[4] = CH;                               // tile_dim1 = CH, tile_dim2 = 0
    g1[5] = PLD;                              // tensor_dim0_stride [31:0]
    g1[6] = 0;                                // dim0_stride hi | dim1_stride lo
    g1[7] = 0;
    const i32x4 z4 = {0, 0, 0, 0};
#if defined(__clang_major__) && (__clang_major__ >= 23)
    const i32x8 z8 = {0, 0, 0, 0, 0, 0, 0, 0};
    __builtin_amdgcn_tensor_load_to_lds(g0, g1, z4, z4, z8, 0);
#else
    __builtin_amdgcn_tensor_load_to_lds(g0, g1, z4, z4, 0);
#endif
}

// ---------------------------------------------------------------------------
// Branch-free fp32 WMMA GEMM:  C[M x N] = A[M x K] * B[K x N], row-major.
// One wave per 16x16 C tile, V_WMMA_F32_16X16X4_F32. Software-pipelined one
// k-step ahead so the wait before each wmma retires at loadcnt<=3 instead of
// draining to zero. Requires K % 16 == 0, lda even, N fully padded.
// ---------------------------------------------------------------------------
__global__ __launch_bounds__(256) void wmma_gemm_f32(
    const float* __restrict__ A, const float* __restrict__ Bm, float* __restrict__ C,
    int Ntiles, int K, int lda, int ldb, int ldc)
{
    const int wave = threadIdx.x >> 5;
    const int lane = threadIdx.x & 31;
    const int tile = blockIdx.x * 8 + wave;   // 8 consecutive tiles share the M-row
    const int tm = tile / Ntiles;
    const int tn = tile % Ntiles;
    const int m0 = tm * 16;
    const int n0 = tn * 16;
    const int half = lane >> 4;
    const int l15  = lane & 15;
    const int col  = n0 + l15;

    const float* __restrict__ ap = A + (size_t)(m0 + l15) * lda + 2 * half;
    const float* __restrict__ bp = Bm + (size_t)(2 * half) * ldb + col;
    const size_t ldbs = (size_t)ldb;
    const size_t ldb4 = 4 * ldbs;

    v2f a = *(const v2f*)ap;                  // prologue: k = 0 operands
    v2f b;
    b.x = bp[0];
    b.y = bp[ldbs];

    v8f c = {};
    #pragma unroll 4
    for (int k = 4; k < K; k += 4) {
        ap += 4;
        bp += ldb4;
        const v2f an = *(const v2f*)ap;       // issue next-step loads first
        v2f bn;
        bn.x = bp[0];
        bn.y = bp[ldbs];
        c = __builtin_amdgcn_wmma_f32_16x16x4_f32(
                false, a, false, b, (short)0, c, false, false);
        a = an;
        b = bn;
    }
    c = __builtin_amdgcn_wmma_f32_16x16x4_f32(
            false, a, false, b, (short)0, c, false, false);

    float* __restrict__ cp = C + (size_t)(m0 + 8 * half) * ldc + col;
    #pragma unroll
    for (int r = 0; r < 8; ++r)
        cp[(size_t)r * ldc] = c[r];
}

// ---------------------------------------------------------------------------
// Zero-pad xproj_w (DI x 129) into (DI x PLD).
// ---------------------------------------------------------------------------
__global__ __launch_bounds__(256) void pad_xproj_kernel(
    const float* __restrict__ w, float* __restrict__ wp)
{
    const int idx = blockIdx.x * 256 + threadIdx.x;   // [0, DI*PLD)
    if (idx >= DI * PLD) return;
    const int row = idx / PLD;
    const int cl  = idx % PLD;
    wp[idx] = (cl < NP) ? w[row * NP + cl] : 0.0f;
}

// ---------------------------------------------------------------------------
// Depthwise causal conv(DC=4) + bias + SiLU.
// ---------------------------------------------------------------------------
__global__ __launch_bounds__(256) void conv_silu_kernel(
    const float* __restrict__ xz, const float* __restrict__ conv_w,
    const float* __restrict__ conv_b, float* __restrict__ xc)
{
    const int idx = blockIdx.x * 256 + threadIdx.x;   // [0, BL*DI)
    const int d   = idx % DI;
    const int row = idx / DI;                         // b*L + t
    const int t   = row % L_;

    float acc = conv_b[d];
    #pragma unroll
    for (int j = 0; j < DC; ++j) {
        const int tt = t - (DC - 1) + j;
        if (tt >= 0)
            acc += xz[(size_t)(row - (DC - 1) + j) * (2 * DI) + d] * conv_w[d * DC + j];
    }
    xc[(size_t)row * DI + d] = silu_f(acc);
}

// ---------------------------------------------------------------------------
// Selective scan with TDM-staged p rows.
// One lane owns one (b, d) channel: h[64] and A-row[64] in registers. The
// per-timestep (B_t, C_t, dt) rows are DMA'd 16 timesteps at a time into a
// double-buffered LDS tile by the Tensor Data Mover (wave 0 issues; TENSORcnt
// waits; block barrier publishes), overlapping DMA with the serial recurrence.
// ---------------------------------------------------------------------------
__global__ __launch_bounds__(256) void scan_kernel(
    const float* __restrict__ xc, const float* __restrict__ p,
    const float* __restrict__ A_log, const float* __restrict__ dt_bias,
    float* __restrict__ y)
{
    const int b = blockIdx.x / (DI / 256);
    const int d = (blockIdx.x % (DI / 256)) * 256 + threadIdx.x;

    __shared__ float sP[2][CH * PLD];   // 2 x 9 KB staging buffers

    float Arow[DS], h[DS];
    #pragma unroll
    for (int n = 0; n < DS; ++n) {
        Arow[n] = -expf(A_log[(size_t)d * DS + n]);
        h[n] = 0.0f;
    }
    const float dtb = dt_bias[d];
    const int row_base = b * L_;
    const int NCH = L_ / CH;            // 256 chunks

    if (threadIdx.x < 32) {             // wave 0 issues the DMA (EXEC-independent)
        tdm_load_rows(p + (size_t)row_base * PLD,
                      (unsigned)(uintptr_t)&sP[0][0]);
        tdm_load_rows(p + (size_t)(row_base + CH) * PLD,
                      (unsigned)(uintptr_t)&sP[1][0]);
    }

    for (int ch = 0; ch < NCH; ++ch) {
        const int buf = ch & 1;
        if (threadIdx.x < 32) {
            if (ch + 1 < NCH) __builtin_amdgcn_s_wait_tensorcnt(1);  // chunk ch landed
            else              __builtin_amdgcn_s_wait_tensorcnt(0);
        }
        __syncthreads();                 // publish buffer to all 8 waves

        const float* __restrict__ sp = &sP[buf][0];
        for (int tt = 0; tt < CH; ++tt) {
            const int row = row_base + ch * CH + tt;
            const float xt = xc[(size_t)row * DI + d];
            const float dt = softplus_f(sp[tt * PLD + 128] + dtb);
            float yv = 0.0f;
            #pragma unroll
            for (int n = 0; n < DS; ++n) {
                h[n] = expf(Arow[n] * dt) * h[n] + dt * sp[tt * PLD + n] * xt;
                yv += sp[tt * PLD + 64 + n] * h[n];
            }
            y[(size_t)row * DI + d] = yv;
        }
        __syncthreads();                 // everyone done reading buf
        if ((threadIdx.x < 32) && (ch + 2 < NCH)) {
            tdm_load_rows(p + (size_t)(row_base + (ch + 2) * CH) * PLD,
                          (unsigned)(uintptr_t)&sP[buf][0]);   // refill freed buffer
        }
    }
}

// ---------------------------------------------------------------------------
// Gate: y <- (y + D*xc) * silu(z), in place. z is xz columns [DI, 2*DI).
// ---------------------------------------------------------------------------
__global__ __launch_bounds__(256) void gate_kernel(
    const float* __restrict__ xz, const float* __restrict__ xc,
    const float* __restrict__ Dp, float* __restrict__ y)
{
    const int idx = blockIdx.x * 256 + threadIdx.x;   // [0, BL*DI)
    const int d   = idx % DI;
    const int row = idx / DI;
    const float z = xz[(size_t)row * (2 * DI) + DI + d];
    y[idx] = (y[idx] + Dp[d] * xc[idx]) * silu_f(z);
}

// ---------------------------------------------------------------------------
extern "C" void kernel_launch(void* const* d_in, const int* in_sizes, int n_in,
                              void* d_out, int out_size, void* d_ws, size_t ws_size,
                              hipStream_t stream)
{
    const float* x       = (const float*)d_in[0];
    const float* in_w    = (const float*)d_in[1];
    const float* conv_w  = (const float*)d_in[2];
    const float* conv_b  = (const float*)d_in[3];
    const float* xproj_w = (const float*)d_in[4];
    const float* A_log   = (const float*)d_in[5];
    const float* dt_bias = (const float*)d_in[6];
    const float* Dp      = (const float*)d_in[7];
    const float* out_w   = (const float*)d_in[8];
    float* out = (float*)d_out;

    float* ws = (float*)d_ws;
    float* xz = ws;                          // BL * 2*DI
    float* xc = xz + (size_t)BL * 2 * DI;    // BL * DI
    float* p  = xc + (size_t)BL * DI;        // BL * PLD
    float* y  = p  + (size_t)BL * PLD;       // BL * DI
    float* wp = y  + (size_t)BL * DI;        // DI * PLD (padded xproj_w)

    // 0) zero-pad xproj_w to DI x PLD
    pad_xproj_kernel<<<(DI * PLD + 255) / 256, 256, 0, stream>>>(xproj_w, wp);

    // 1) xz = x @ in_w                 M=8192, K=768,  N=3072
    {
        const int Nt = (2 * DI) / 16;                 // 192
        const int tiles = (BL / 16) * Nt;             // 98304
        wmma_gemm_f32<<<tiles / 8, 256, 0, stream>>>(
            x, in_w, xz, Nt, DM, DM, 2 * DI, 2 * DI);
    }

    // 2) xc = silu(depthwise_conv(x_ssm) + b)
    conv_silu_kernel<<<(BL * DI) / 256, 256, 0, stream>>>(xz, conv_w, conv_b, xc);

    // 3) p = xc @ wp                   M=8192, K=1536, N=144 (129 valid)
    {
        const int Nt = PLD / 16;                      // 9
        const int tiles = (BL / 16) * Nt;             // 4608
        wmma_gemm_f32<<<tiles / 8, 256, 0, stream>>>(
            xc, wp, p, Nt, DI, DI, PLD, PLD);
    }

    // 4) selective scan -> y  (TDM double-buffered p rows)
    scan_kernel<<<B_ * (DI / 256), 256, 0, stream>>>(xc, p, A_log, dt_bias, y);

    // 5) y <- (y + D*xc) * silu(z)
    gate_kernel<<<(BL * DI) / 256, 256, 0, stream>>>(xz, xc, Dp, y);

    // 6) out = y @ out_w               M=8192, K=1536, N=768
    {
        const int Nt = DM / 16;                       // 48
        const int tiles = (BL / 16) * Nt;             // 24576
        wmma_gemm_f32<<<tiles / 8, 256, 0, stream>>>(
            y, out_w, out, Nt, DI, DI, DM, DM);
    }
}